// Encoder_23424751632448
// MI455X (gfx1250) — compile-verified
//
#include <hip/hip_runtime.h>

typedef __attribute__((ext_vector_type(2))) float f32x2;
typedef __attribute__((ext_vector_type(8))) float f32x8;

#define VOCAB 32000
#define EMB   512
#define HID   1024
#define GATES 4096   // 4*HID
#define BATCH 32
#define SEQ   512

__global__ __launch_bounds__(256) void zero_ws_kernel(float* __restrict__ p, int n) {
    int i = blockIdx.x * blockDim.x + threadIdx.x;
    if (i < n) p[i] = 0.0f;
}

__device__ __forceinline__ float sigmoidf_dev(float x) {
    return 1.0f / (1.0f + __expf(-x));
}

// One LSTM timestep. Grid: HID/16 = 64 blocks, each owns hidden columns
// [n0, n0+16). Block: 256 threads = 8 wave32; wave = (gate 0..3, batch-half 0..1).
// Each wave accumulates a 16x16 fp32 tile of pre-activation gate values with
// V_WMMA_F32_16X16X4_F32 over K = EMB (embedded input) + HID (h_prev).
__global__ __launch_bounds__(256) void lstm_step_kernel(
    const int*   __restrict__ src,
    const float* __restrict__ emb,
    const float* __restrict__ W_ih,
    const float* __restrict__ W_hh,
    const float* __restrict__ b_ih,
    const float* __restrict__ b_hh,
    const float* __restrict__ h_prev, long h_prev_bstride,
    float*       __restrict__ c_state,
    float*       __restrict__ h_out, long h_out_bstride,
    int t)
{
    __shared__ float lds_g[4][BATCH][16];   // i,f,g,o pre-activation tiles

    const int tid  = threadIdx.x;
    const int lane = tid & 31;
    const int wave = tid >> 5;      // 0..7
    const int gate = wave >> 1;     // 0..3 -> i,f,g,o (jnp.split order)
    const int mt   = wave & 1;      // batch half: rows mt*16 .. mt*16+15
    const int l16  = lane & 15;
    const int hi   = lane >> 4;     // 0: K=k0,k0+1   1: K=k0+2,k0+3

    const int n0 = blockIdx.x * 16;       // hidden-tile base
    const int N  = n0 + l16;              // hidden column this lane owns (B/D)
    const int M  = mt * 16 + l16;         // batch row this lane supplies (A)
    const int gcol = gate * HID + N;      // column in the 4H gate matrix

    f32x8 acc = {0.f, 0.f, 0.f, 0.f, 0.f, 0.f, 0.f, 0.f};

    // ---- input part: A = embedding[src[M][t]] (16 x EMB), B = W_ih[:, gcol]
    {
        const long row = (long)src[M * SEQ + t];
        const float* a_ptr = emb + row * (long)EMB;
        #pragma unroll 4
        for (int k0 = 0; k0 < EMB; k0 += 4) {
            const int kb = k0 + 2 * hi;
            f32x2 a = *(const f32x2*)(a_ptr + kb);          // 8B contiguous
            f32x2 b;
            b.x = W_ih[(long)kb * GATES + gcol];
            b.y = W_ih[(long)(kb + 1) * GATES + gcol];
            acc = __builtin_amdgcn_wmma_f32_16x16x4_f32(
                false, a, false, b, (short)0, acc, false, false);
        }
    }

    // ---- recurrent part: A = h_prev (16 x HID), B = W_hh[:, gcol]
    {
        const float* a_ptr = h_prev + (long)M * h_prev_bstride;
        #pragma unroll 4
        for (int k0 = 0; k0 < HID; k0 += 4) {
            const int kb = k0 + 2 * hi;
            f32x2 a = *(const f32x2*)(a_ptr + kb);
            f32x2 b;
            b.x = W_hh[(long)kb * GATES + gcol];
            b.y = W_hh[(long)(kb + 1) * GATES + gcol];
            acc = __builtin_amdgcn_wmma_f32_16x16x4_f32(
                false, a, false, b, (short)0, acc, false, false);
        }
    }

    // Per-column bias (b_ih + b_hh), same for every batch row.
    const float bias = b_ih[gcol] + b_hh[gcol];

    // D-matrix layout: VGPR r holds row (r + 8*hi) of the 16-row tile, col l16.
    #pragma unroll
    for (int r = 0; r < 8; ++r) {
        const int brow = mt * 16 + hi * 8 + r;    // batch index 0..31
        lds_g[gate][brow][l16] = acc[r] + bias;
    }
    __syncthreads();

    // Pointwise LSTM cell update: 32 batch x 16 cols = 512 values, 2 per thread.
    #pragma unroll
    for (int idx = tid; idx < BATCH * 16; idx += 256) {
        const int b = idx >> 4;
        const int n = idx & 15;
        const float ig = sigmoidf_dev(lds_g[0][b][n]);
        const float fg = sigmoidf_dev(lds_g[1][b][n]);
        const float gg = tanhf(lds_g[2][b][n]);
        const float og = sigmoidf_dev(lds_g[3][b][n]);
        const long cidx = (long)b * HID + n0 + n;
        const float c_new = fg * c_state[cidx] + ig * gg;
        c_state[cidx] = c_new;
        h_out[(long)b * h_out_bstride + n0 + n] = og * tanhf(c_new);
    }
}

extern "C" void kernel_launch(void* const* d_in, const int* in_sizes, int n_in,
                              void* d_out, int out_size, void* d_ws, size_t ws_size,
                              hipStream_t stream) {
    const int*   src  = (const int*)  d_in[0];   // [BATCH][SEQ]
    const float* emb  = (const float*)d_in[1];   // [VOCAB][EMB]
    const float* W_ih = (const float*)d_in[2];   // [EMB][4H]
    const float* W_hh = (const float*)d_in[3];   // [HID][4H]
    const float* b_ih = (const float*)d_in[4];   // [4H]
    const float* b_hh = (const float*)d_in[5];   // [4H]
    float* out = (float*)d_out;                  // [BATCH][SEQ][HID]

    float* c_state = (float*)d_ws;               // [BATCH][HID]
    float* h0      = c_state + BATCH * HID;      // [BATCH][HID] zeros

    // Re-initialize carried state every call (deterministic across replays).
    const int n_init = 2 * BATCH * HID;
    zero_ws_kernel<<<(n_init + 255) / 256, 256, 0, stream>>>(c_state, n_init);

    for (int t = 0; t < SEQ; ++t) {
        const float* h_prev = (t == 0) ? h0 : (out + (long)(t - 1) * HID);
        const long   hstr   = (t == 0) ? (long)HID : (long)SEQ * HID;
        lstm_step_kernel<<<HID / 16, 256, 0, stream>>>(
            src, emb, W_ih, W_hh, b_ih, b_hh,
            h_prev, hstr, c_state,
            out + (long)t * HID, (long)SEQ * HID, t);
    }
}